// VQ_34548716929075
// MI455X (gfx1250) — compile-verified
//
#include <hip/hip_runtime.h>
#include <math.h>

#define K_CODES   512
#define DIM       64
#define TOK_BLK   128          // tokens per block = two contiguous (b,h,:) rows
#define THREADS   128          // 4 waves; each wave owns 2 M-tiles (32 tokens)
#define CHUNK     64           // codebook rows staged in LDS per iteration
#define XS_STRIDE 65           // +1 pad -> conflict-free strided LDS reads
#define ES_STRIDE 65
#define N_TOKENS  131072
#define OUT_ELEMS 8388608      // 32*64*64*64

typedef __attribute__((ext_vector_type(2))) float v2f;
typedef __attribute__((ext_vector_type(8))) float v8f;

__global__ void vq_zero(int* counts, float* sse) {
    int t = threadIdx.x;
    if (t < K_CODES) counts[t] = 0;
    if (t == 0) *sse = 0.0f;
}

__global__ __launch_bounds__(THREADS)
void vq_main(const float* __restrict__ x, const float* __restrict__ emb,
             float* __restrict__ out, int* __restrict__ counts,
             float* __restrict__ sse) {
    __shared__ float xs[TOK_BLK * XS_STRIDE];
    __shared__ float es[CHUNK * ES_STRIDE];
    __shared__ float enorm_s[CHUNK];
    __shared__ int   idx_s[TOK_BLK];

    const int t  = threadIdx.x;
    const int n0 = blockIdx.x * TOK_BLK;     // 128 consecutive tokens, same b
    const int b  = n0 >> 12;
    const int h0 = (n0 & 4095) >> 6;
    // rows h0,h0+1 are memory-contiguous: addr = xbase + d*4096 + tok
    const long xbase = (long)b * 262144 + (long)h0 * 64;

    // ---- stage x tile: 128 tokens x 64 dims, coalesced per-d rows ----
    for (int j = 0; j < 64; ++j) {
        int f = j * THREADS + t;             // 0..8191
        int d = f >> 7, tok = f & 127;
        xs[tok * XS_STRIDE + d] = x[xbase + (long)d * 4096 + tok];
    }
    __syncthreads();

    const int lane  = t & 31;
    const int wave  = t >> 5;
    const int l16   = lane & 15;             // M for A-frag, N for B/C/D-frag
    const int khalf = lane >> 4;
    const int tw0   = wave * 32;             // this wave's 32-token base

    // ---- hoist A-fragments for both M-tiles into registers (invariant) ----
    v2f afrag[2][DIM / 4];
#pragma unroll
    for (int tile = 0; tile < 2; ++tile) {
        int row = (tw0 + tile * 16 + l16) * XS_STRIDE;
#pragma unroll
        for (int kk = 0; kk < DIM / 4; ++kk) {
            int k = kk * 4 + khalf * 2;
            afrag[tile][kk].x = xs[row + k];
            afrag[tile][kk].y = xs[row + k + 1];
        }
    }

    float bestd[2][8];
    int   bestn[2][8];
#pragma unroll
    for (int tile = 0; tile < 2; ++tile)
#pragma unroll
        for (int v = 0; v < 8; ++v) { bestd[tile][v] = 3.4e38f; bestn[tile][v] = 0; }

    for (int c = 0; c < K_CODES / CHUNK; ++c) {
        __syncthreads();                     // previous chunk fully consumed
        // ---- stage codebook chunk: 64 codes x 64 dims ----
        const long ebase = (long)c * CHUNK * DIM;
        for (int j = 0; j < 32; ++j) {
            int f = j * THREADS + t;         // 0..4095
            es[(f >> 6) * ES_STRIDE + (f & 63)] = emb[ebase + f];
        }
        __syncthreads();
        if (t < CHUNK) {                     // per-code squared norms
            float s = 0.0f;
            for (int d = 0; d < DIM; ++d) { float e = es[t * ES_STRIDE + d]; s += e * e; }
            enorm_s[t] = s;
        }
        __syncthreads();

        // ---- 16x16 distance tiles: each B-fragment feeds 2 WMMAs ----
        for (int nt = 0; nt < CHUNK / 16; ++nt) {
            v8f acc0 = {}, acc1 = {};
            int brow = (nt * 16 + l16) * ES_STRIDE;
#pragma unroll
            for (int kk = 0; kk < DIM / 4; ++kk) {
                int k = kk * 4 + khalf * 2;
                v2f bf;
                bf.x = es[brow + k];
                bf.y = es[brow + k + 1];
                acc0 = __builtin_amdgcn_wmma_f32_16x16x4_f32(
                           false, afrag[0][kk], false, bf, (short)0, acc0, false, false);
                acc1 = __builtin_amdgcn_wmma_f32_16x16x4_f32(
                           false, afrag[1][kk], false, bf, (short)0, acc1, false, false);
            }
            // argmin over (||e||^2 - 2*x.e); token-norm constant per row -> dropped
            float en    = enorm_s[nt * 16 + l16];
            int   nglob = c * CHUNK + nt * 16 + l16;
#pragma unroll
            for (int v = 0; v < 8; ++v) {    // C/D layout: m = v + 8*khalf, n = l16
                float d0 = en - 2.0f * acc0[v];
                if (d0 < bestd[0][v] || (d0 == bestd[0][v] && nglob < bestn[0][v])) {
                    bestd[0][v] = d0; bestn[0][v] = nglob;
                }
                float d1 = en - 2.0f * acc1[v];
                if (d1 < bestd[1][v] || (d1 == bestd[1][v] && nglob < bestn[1][v])) {
                    bestd[1][v] = d1; bestn[1][v] = nglob;
                }
            }
        }
    }

    // ---- min-reduce across the 16 n-lanes inside each half-wave (wave32) ----
#pragma unroll
    for (int mask = 1; mask < 16; mask <<= 1) {
#pragma unroll
        for (int tile = 0; tile < 2; ++tile) {
#pragma unroll
            for (int v = 0; v < 8; ++v) {
                float od = __shfl_xor(bestd[tile][v], mask, 32);
                int   on = __shfl_xor(bestn[tile][v], mask, 32);
                if (od < bestd[tile][v] ||
                    (od == bestd[tile][v] && on < bestn[tile][v])) {
                    bestd[tile][v] = od; bestn[tile][v] = on;
                }
            }
        }
    }
    if (l16 == 0) {
#pragma unroll
        for (int tile = 0; tile < 2; ++tile)
#pragma unroll
            for (int v = 0; v < 8; ++v)      // token m = v + 8*khalf
                idx_s[tw0 + tile * 16 + khalf * 8 + v] = bestn[tile][v];
    }
    __syncthreads();

    // ---- fused epilogue: gather, NCHW store, SSE, histogram ----
    float lsum = 0.0f;
    for (int j = 0; j < 64; ++j) {
        int f = j * THREADS + t;
        int d = f >> 7, tok = f & 127;
        int code = idx_s[tok];
        float q  = emb[code * DIM + d];      // 128KB table -> L2-resident
        float xv = xs[tok * XS_STRIDE + d];
        out[1 + xbase + (long)d * 4096 + tok] = q;  // out tensor starts at d_out[1]
        float df = q - xv;
        lsum += df * df;
    }
#pragma unroll
    for (int mask = 1; mask < 32; mask <<= 1)
        lsum += __shfl_xor(lsum, mask, 32);
    if (lane == 0) atomicAdd(sse, lsum);
    if (t < TOK_BLK) atomicAdd(&counts[idx_s[t]], 1);
}

__global__ void vq_finalize(const int* __restrict__ counts,
                            const float* __restrict__ sse,
                            float* __restrict__ out) {
    __shared__ float red[K_CODES];
    int t = threadIdx.x;
    float p = (float)counts[t] * (1.0f / (float)N_TOKENS);
    red[t] = p * logf(p + 1e-10f);
    __syncthreads();
    for (int s = K_CODES / 2; s > 0; s >>= 1) {
        if (t < s) red[t] += red[t + s];
        __syncthreads();
    }
    if (t == 0) {
        // loss = (1 + BETA) * MSE ; both loss terms equal in forward value
        out[0]             = (*sse) * (1.25f / (float)OUT_ELEMS);
        out[1 + OUT_ELEMS] = expf(-red[0]);
    }
}

extern "C" void kernel_launch(void* const* d_in, const int* in_sizes, int n_in,
                              void* d_out, int out_size, void* d_ws, size_t ws_size,
                              hipStream_t stream) {
    const float* x   = (const float*)d_in[0];
    const float* emb = (const float*)d_in[1];
    float* out  = (float*)d_out;
    int*   cnts = (int*)d_ws;
    float* sse  = (float*)((char*)d_ws + K_CODES * sizeof(int));

    vq_zero<<<1, K_CODES, 0, stream>>>(cnts, sse);
    vq_main<<<N_TOKENS / TOK_BLK, THREADS, 0, stream>>>(x, emb, out, cnts, sse);
    vq_finalize<<<1, K_CODES, 0, stream>>>(cnts, sse, out);
}